// GDLoss_v1_80324478370041
// MI455X (gfx1250) — compile-verified
//
#include <hip/hip_runtime.h>
#include <cstdint>

// ---------------------------------------------------------------------------
// KLD rotated-box loss, N=4M pairs. Memory-bound target: 176 MB @ 23.3 TB/s
// ~ 7.5 us floor. Data path: CDNA5 GLOBAL_LOAD_ASYNC_TO_LDS_B128 staging
// (ASYNCcnt), stride-5 LDS reads (gcd(5,64)=1 -> conflict-free), coalesced
// b32 store. Compute path: input r is in [-pi/2, pi/2] by construction, so
// hardware v_sin/v_cos (__sincosf) is exact enough -- avoids the Payne-Hanek
// slow path that would otherwise make this VALU-bound. v_rcp_f32 replaces
// IEEE divides. No GEMM structure exists -> WMMA is not applicable here.
// ---------------------------------------------------------------------------

#define BOXES_PER_BLOCK 256
#define FLOATS_PER_BUF  (BOXES_PER_BLOCK * 5)   // 1280 floats = 5120 B
#define CHUNKS          (FLOATS_PER_BUF / 4)    // 320 x float4 per buffer

// Exact pointer types the gfx1250 async builtin expects (from clang diag):
//   param0: 'int __vector_size__(16) __device__ *'   (addrspace(1), non-const)
//   param1: LDS side -> addrspace(3)
typedef int v4i __attribute__((vector_size(16)));
typedef __attribute__((address_space(1))) v4i g_v4i;
typedef __attribute__((address_space(3))) v4i l_v4i;

// 16B global -> LDS async copy (per-lane; one wave32 instr moves 512B).
__device__ __forceinline__ void async_copy_b128(const float* gsrc, float* ldst) {
#if __has_builtin(__builtin_amdgcn_global_load_async_to_lds_b128)
    __builtin_amdgcn_global_load_async_to_lds_b128(
        (g_v4i*)gsrc, (l_v4i*)ldst, /*imm offset*/0, /*cpol*/0);
#else
    unsigned lds_off = (unsigned)(uintptr_t)(__attribute__((address_space(3))) float*)ldst;
    asm volatile("global_load_async_to_lds_b128 %0, %1, off"
                 :: "v"(lds_off), "v"(gsrc) : "memory");
#endif
}

__device__ __forceinline__ void wait_async_zero() {
#if __has_builtin(__builtin_amdgcn_s_wait_asynccnt)
    __builtin_amdgcn_s_wait_asynccnt(0);
#else
    asm volatile("s_wait_asynccnt 0" ::: "memory");
#endif
}

__device__ __forceinline__ float fast_rcp(float x) {
    return __builtin_amdgcn_rcpf(x);             // v_rcp_f32, ~1 ulp
}

// rbox (x,y,w,h,r) -> Gaussian (mu, {sxx,sxy,syy})
__device__ __forceinline__ void box_to_gauss(const float b[5],
                                             float& mx, float& my,
                                             float& sxx, float& sxy, float& syy) {
    mx = b[0];
    my = b[1];
    float w = fminf(fmaxf(b[2], 1e-7f), 1e7f);
    float h = fminf(fmaxf(b[3], 1e-7f), 1e7f);
    float s, c;
    __sincosf(b[4], &s, &c);    // r in [-pi/2,pi/2]: hw v_sin/v_cos, no range reduction
    float a  = 0.25f * w * w;   // (w/2)^2
    float bb = 0.25f * h * h;   // (h/2)^2
    sxx = a * c * c + bb * s * s;
    syy = a * s * s + bb * c * c;
    sxy = (a - bb) * s * c;
}

__global__ void __launch_bounds__(BOXES_PER_BLOCK)
kld_loss_kernel(const float* __restrict__ pred,
                const float* __restrict__ targ,
                float* __restrict__ out, int n) {
    __shared__ float sp[FLOATS_PER_BUF];
    __shared__ float st[FLOATS_PER_BUF];

    const int t    = threadIdx.x;
    const int base = blockIdx.x * BOXES_PER_BLOCK;
    const int nbox = min(BOXES_PER_BLOCK, n - base);
    const bool full = (nbox == BOXES_PER_BLOCK);   // uniform per block

    if (full) {
        const float* gp = pred + (size_t)base * 5;
        const float* gt = targ + (size_t)base * 5;
        // chunk t (all 256 threads), chunk 256+t (first 64 threads)
        async_copy_b128(gp + 4 * t, sp + 4 * t);
        async_copy_b128(gt + 4 * t, st + 4 * t);
        if (t < (CHUNKS - BOXES_PER_BLOCK)) {      // t < 64
            const int c = BOXES_PER_BLOCK + t;
            async_copy_b128(gp + 4 * c, sp + 4 * c);
            async_copy_b128(gt + 4 * c, st + 4 * c);
        }
        wait_async_zero();                          // own wave's ASYNCcnt -> 0
    }
    __syncthreads();                                // publish LDS across waves

    if (t >= nbox) return;

    float bp[5], bt[5];
    if (full) {
        // stride-5 LDS reads: gcd(5,64)=1 -> bank-conflict free
#pragma unroll
        for (int i = 0; i < 5; ++i) { bp[i] = sp[5 * t + i]; bt[i] = st[5 * t + i]; }
    } else {
        const float* gp = pred + (size_t)(base + t) * 5;
        const float* gt = targ + (size_t)(base + t) * 5;
#pragma unroll
        for (int i = 0; i < 5; ++i) { bp[i] = gp[i]; bt[i] = gt[i]; }
    }

    float mxp, myp, pxx, pxy, pyy;
    float mxt, myt, txx, txy, tyy;
    box_to_gauss(bp, mxp, myp, pxx, pxy, pyy);
    box_to_gauss(bt, mxt, myt, txx, txy, tyy);

    const float dx = mxp - mxt;
    const float dy = myp - myt;

    const float det_t = txx * tyy - txy * txy;
    const float det_p = pxx * pyy - pxy * pxy;
    const float inv_det_t = fast_rcp(det_t);

    // delta^T Sigma_t^{-1} delta
    const float term1 = (dx * dx * tyy - 2.0f * dx * dy * txy + dy * dy * txx) * inv_det_t;
    // trace(Sigma_t^{-1} Sigma_p)
    const float tr    = (tyy * pxx - 2.0f * txy * pxy + txx * pyy) * inv_det_t;

    const float term2 = tr + __logf(det_t * fast_rcp(det_p));
    const float dis   = term1 + term2 - 2.0f;
    const float kl    = fmaxf(dis, 1e-6f);
    const float loss  = 1.0f - fast_rcp(1.0f + log1pf(kl));   // tau = 1.0

    out[base + t] = loss;                                     // coalesced b32
}

extern "C" void kernel_launch(void* const* d_in, const int* in_sizes, int n_in,
                              void* d_out, int out_size, void* d_ws, size_t ws_size,
                              hipStream_t stream) {
    const float* pred = (const float*)d_in[0];
    const float* targ = (const float*)d_in[1];
    float* out = (float*)d_out;
    const int n = in_sizes[0] / 5;                 // 4,000,000
    const int grid = (n + BOXES_PER_BLOCK - 1) / BOXES_PER_BLOCK;
    kld_loss_kernel<<<grid, BOXES_PER_BLOCK, 0, stream>>>(pred, targ, out, n);
}